// Mask_83099027243166
// MI455X (gfx1250) — compile-verified
//
#include <hip/hip_runtime.h>
#include <hip/hip_bf16.h>
#include <math.h>

typedef float v4f __attribute__((ext_vector_type(4)));
typedef float v2f __attribute__((ext_vector_type(2)));

#define TPB   256
#define ITERS 4

// CDNA5 async global->LDS copy (per-lane 16B), tracked by ASYNCcnt.
// NT temporal hint: x is streamed exactly once (253 MB) — keep it from
// evicting the L2-resident 1 MB coefficient table that is re-read 512x.
__device__ __forceinline__ void async_ld_f4_nt(const v4f* gptr, const v4f* lptr) {
  unsigned long long ga = (unsigned long long)gptr;           // 64-bit global addr -> VGPR pair
  unsigned           la = (unsigned)(unsigned long long)lptr; // low 32 bits = LDS offset
  asm volatile("global_load_async_to_lds_b128 %0, %1, off th:TH_LOAD_NT"
               :: "v"(la), "v"(ga)
               : "memory");
}

// ---------------------------------------------------------------------------
// Setup: window product -> softmax(500) -> pairwise alpha max -> fused coef
// table coef[p] = { alpha[p], (1 - alpha[p]/max_alpha) * noise[p] * 0.1 }.
// Single block, 512 threads (16 wave32) — negligible vs the streaming kernel.
// ---------------------------------------------------------------------------
__global__ __launch_bounds__(512)
void mask_setup(const float* __restrict__ weight,
                const float* __restrict__ noise,
                const int*   __restrict__ r1,
                const int*   __restrict__ r2,
                v2f*         __restrict__ coef,
                int n_res, int input_size)
{
  __shared__ float s[512];
  __shared__ float red[512];
  const int tid = threadIdx.x;

  float wv = 0.0f;
  if (tid < n_res)
    wv = weight[tid] * weight[tid + 1] * weight[tid + 2] * weight[tid + 3];

  // softmax max-reduce
  red[tid] = (tid < n_res) ? wv : -INFINITY;
  __syncthreads();
  for (int off = 256; off > 0; off >>= 1) {
    if (tid < off) red[tid] = fmaxf(red[tid], red[tid + off]);
    __syncthreads();
  }
  const float wmax = red[0];
  __syncthreads();

  // exp + sum-reduce
  const float e = (tid < n_res) ? expf(wv - wmax) : 0.0f;
  red[tid] = e;
  __syncthreads();
  for (int off = 256; off > 0; off >>= 1) {
    if (tid < off) red[tid] += red[tid + off];
    __syncthreads();
  }
  const float denom = red[0];
  if (tid < n_res) s[tid] = e / denom;
  __syncthreads();

  // max over pair products s[i]*s[j]  (ratio alpha/max is K-independent)
  float lmax = 0.0f;
  for (int p = tid; p < input_size; p += 512)
    lmax = fmaxf(lmax, s[r1[p]] * s[r2[p]]);
  red[tid] = lmax;
  __syncthreads();
  for (int off = 256; off > 0; off >>= 1) {
    if (tid < off) red[tid] = fmaxf(red[tid], red[tid + off]);
    __syncthreads();
  }
  const float amax = red[0];
  const float K = (float)n_res * (float)n_res;

  for (int p = tid; p < input_size; p += 512) {
    const float a = s[r1[p]] * s[r2[p]];
    v2f c;
    c.x = a * K;                                   // alpha
    c.y = (1.0f - a / amax) * noise[p] * 0.1f;     // additive noise term
    coef[p] = c;
  }
}

// ---------------------------------------------------------------------------
// Streaming kernel: out[n] = x[n]*c0[n%L] + c1[n%L], 16B/lane, async-staged
// x tiles through LDS (double buffer, ASYNCcnt depth 2), NT stores for out.
// ---------------------------------------------------------------------------
__global__ __launch_bounds__(TPB)
void mask_main(const float* __restrict__ x,
               const v2f*   __restrict__ coef,
               float*       __restrict__ out,
               unsigned nvec, unsigned L, unsigned total)
{
  __shared__ v4f buf[2][TPB];
  const unsigned tid    = threadIdx.x;
  const unsigned stride = gridDim.x * TPB;
  const unsigned idx0   = blockIdx.x * TPB + tid;
  const v4f* __restrict__ x4 = (const v4f*)x;
  v4f*       __restrict__ o4 = (v4f*)out;

  // Prologue: stage iteration 0. Clamp (not predicate) out-of-range lanes so
  // every wave issues exactly one async op per stage -> ASYNCcnt bookkeeping
  // is deterministic regardless of EXEC.
  {
    const unsigned i0 = (idx0 < nvec) ? idx0 : 0u;
    async_ld_f4_nt(x4 + i0, &buf[0][tid]);
  }

#pragma unroll
  for (int i = 0; i < ITERS; ++i) {
    const unsigned cur = idx0 + (unsigned)i * stride;

    if (i + 1 < ITERS) {
      unsigned nxt = idx0 + (unsigned)(i + 1) * stride;
      if (nxt >= nvec) nxt = 0u;
      async_ld_f4_nt(x4 + nxt, &buf[(i + 1) & 1][tid]);
      asm volatile("s_wait_asynccnt 0x1" ::: "memory"); // stage i complete
    } else {
      asm volatile("s_wait_asynccnt 0x0" ::: "memory"); // last stage complete
    }

    if (cur < nvec) {
      const v4f xv = buf[i & 1][tid];      // ds_load_b128 from staged tile
      const unsigned e0 = cur * 4u;
      unsigned p0 = e0 % L;
      unsigned p1 = p0 + 1u; if (p1 >= L) p1 -= L;
      unsigned p2 = p1 + 1u; if (p2 >= L) p2 -= L;
      unsigned p3 = p2 + 1u; if (p3 >= L) p3 -= L;
      const v2f ca = coef[p0];             // L2-resident 1MB table (RT policy)
      const v2f cb = coef[p1];
      const v2f cc = coef[p2];
      const v2f cd = coef[p3];
      v4f o;
      o.x = fmaf(xv.x, ca.x, ca.y);
      o.y = fmaf(xv.y, cb.x, cb.y);
      o.z = fmaf(xv.z, cc.x, cc.y);
      o.w = fmaf(xv.w, cd.x, cd.y);
      __builtin_nontemporal_store(o, &o4[cur]);  // write stream bypasses caches
    }
  }

  // Scalar tail if total % 4 != 0 (not the case for BATCH=512, but be exact).
  if (blockIdx.x == 0 && tid == 0) {
    for (unsigned e = nvec * 4u; e < total; ++e) {
      const unsigned p = e % L;
      const v2f c = coef[p];
      out[e] = fmaf(x[e], c.x, c.y);
    }
  }
}

extern "C" void kernel_launch(void* const* d_in, const int* in_sizes, int n_in,
                              void* d_out, int out_size, void* d_ws, size_t ws_size,
                              hipStream_t stream) {
  const float* x      = (const float*)d_in[0];
  const float* weight = (const float*)d_in[1];
  const float* noise  = (const float*)d_in[2];
  const int*   r1     = (const int*)d_in[3];
  const int*   r2     = (const int*)d_in[4];
  float*       out    = (float*)d_out;

  const int n_res      = in_sizes[1] - 3;        // PATCHSIZE - 1
  const int input_size = in_sizes[3];            // len(residues_1)
  const unsigned total = (unsigned)in_sizes[0];  // BATCH * input_size
  const unsigned L     = (unsigned)input_size;
  const unsigned nvec  = total / 4u;

  v2f* coef = (v2f*)d_ws;                        // input_size * 8 bytes (~1 MB)

  mask_setup<<<1, 512, 0, stream>>>(weight, noise, r1, r2, coef, n_res, input_size);

  const unsigned work   = TPB * ITERS;           // vec4 elems per block
  unsigned blocks = (nvec + work - 1u) / work;
  if (blocks == 0u) blocks = 1u;
  mask_main<<<blocks, TPB, 0, stream>>>(x, coef, out, nvec, L, total);
}